// Net_1503238553644
// MI455X (gfx1250) — compile-verified
//
#include <hip/hip_runtime.h>
#include <math.h>

typedef float v2f __attribute__((ext_vector_type(2)));
typedef float v8f __attribute__((ext_vector_type(8)));

#define __AS1 __attribute__((address_space(1)))
#define __AS3 __attribute__((address_space(3)))

#if defined(__AMDGCN__) && __has_builtin(__builtin_amdgcn_global_load_async_to_lds_b32) && \
    __has_builtin(__builtin_amdgcn_s_wait_asynccnt)
#define USE_ASYNC_LDS 1
#else
#define USE_ASYNC_LDS 0
#endif

// ---------------------------------------------------------------------------
// Utility kernels
// ---------------------------------------------------------------------------
__global__ void zero_f32(float* p, size_t n) {
    size_t i = (size_t)blockIdx.x * blockDim.x + threadIdx.x;
    size_t stride = (size_t)gridDim.x * blockDim.x;
    for (; i < n; i += stride) p[i] = 0.f;
}

__global__ void degree_kernel(const int* __restrict__ ei, float* __restrict__ deg, int E) {
    int i = blockIdx.x * blockDim.x + threadIdx.x;
    if (i < E) atomicAdd(&deg[ei[i]], 1.f);   // src = ei[0..E)
}

__global__ void dinv_kernel(float* deg, int N) {
    int i = blockIdx.x * blockDim.x + threadIdx.x;
    if (i < N) {
        float d = deg[i];
        deg[i] = (d > 0.f) ? rsqrtf(d) : 0.f;   // max(d,1)==d when d>0 (counts)
    }
}

__global__ void norm_kernel(const int* __restrict__ ei, const float* __restrict__ dinv,
                            float* __restrict__ norm, int E) {
    int e = blockIdx.x * blockDim.x + threadIdx.x;
    if (e < E) norm[e] = -dinv[ei[e]] * dinv[ei[E + e]];
}

// out[dst,:] += norm[e] * feat[src,:]  — one wave32 per edge, lanes stride features
__global__ void scatter_axpy(const float* __restrict__ feat, const int* __restrict__ ei,
                             const float* __restrict__ norm, float* __restrict__ out,
                             int E, int F) {
    int lane = threadIdx.x & 31;
    int gwarp = blockIdx.x * (blockDim.x >> 5) + (threadIdx.x >> 5);
    int nwarp = gridDim.x * (blockDim.x >> 5);
    for (int e = gwarp; e < E; e += nwarp) {
        int s = ei[e];
        int d = ei[E + e];
        float nv = norm[e];
        const float* fs = feat + (size_t)s * F;
        float* od = out + (size_t)d * F;
        for (int f = lane; f < F; f += 32)
            atomicAdd(&od[f], nv * fs[f]);
    }
}

// width-1 scatter for conv3
__global__ void scatter1(const float* __restrict__ y1, const int* __restrict__ ei,
                         const float* __restrict__ norm, float* __restrict__ out, int E) {
    int e = blockIdx.x * blockDim.x + threadIdx.x;
    int stride = gridDim.x * blockDim.x;
    for (; e < E; e += stride)
        atomicAdd(&out[ei[E + e]], norm[e] * y1[ei[e]]);
}

// ---------------------------------------------------------------------------
// f32 WMMA GEMM with LDS-staged B panel.
//   C[M,N] = epilogue( A0[M,K0]@B0[K0,N] (+ A1[M,K1]@B1[K1,N]) )
// Block = 256 threads = 8 wave32s -> 128x64 output tile (wave w: rows m0..m0+15).
// B panel (Kx64) staged chunk-wise (64 K-rows = 16.6KB, stride 65 -> no bank
// conflicts between half-waves) via async global->LDS when available.
// OOB rows/cols use CLAMPED addresses; garbage only reaches rows/cols the
// guarded epilogue never stores, so all loads are unconditional.
// epilogue: v = acc + bias[col]; if(pre) v+=pre; if(relu) v=max(v,0); if(post) v+=post
// ---------------------------------------------------------------------------
#define BK 64
#define LDB 65   // padded LDS row stride (floats)

__device__ __forceinline__ void stage_b_panel(const float* __restrict__ B, int k0, int KC,
                                              int n0, int N, float* sB) {
    // cooperative: 256 threads stage KC x 64 elements (col-clamped addresses)
    for (int i = threadIdx.x; i < KC * 64; i += 256) {
        int k = i >> 6;
        int c = i & 63;
        int col = n0 + c;
        int colc = (col < N) ? col : (N - 1);
        const float* g = B + (size_t)(k0 + k) * N + colc;
        float* l = &sB[k * LDB + c];
#if USE_ASYNC_LDS
        __builtin_amdgcn_global_load_async_to_lds_b32(
            (__AS1 int*)(void*)g, (__AS3 int*)(void*)l, 0, 0);
#else
        *l = *g;
#endif
    }
}

__device__ __forceinline__ void stage_wait() {
#if USE_ASYNC_LDS
    __builtin_amdgcn_s_wait_asynccnt(0);
#endif
    __syncthreads();
}

__global__ __launch_bounds__(256) void gemm_wmma_f32(
    const float* __restrict__ A0, const float* __restrict__ B0, int K0,
    const float* __restrict__ A1, const float* __restrict__ B1, int K1,
    const float* __restrict__ bias, const float* __restrict__ pre,
    const float* __restrict__ post, float* __restrict__ C,
    int M, int N, int relu)
{
    __shared__ float sB[BK * LDB];

    const int lane = threadIdx.x & 31;
    const int wid  = threadIdx.x >> 5;               // 0..7
    const int m0 = blockIdx.y * 128 + wid * 16;
    const int n0 = blockIdx.x * 64;
    const int mr   = lane & 15;
    const int half = lane >> 4;                      // 0 or 1
    const int krow = half * 2;                       // K-pair base per ISA A/B layout

    const int rowW = m0 + mr;
    const int rowC = (rowW < M) ? rowW : (M - 1);    // clamped A row

    v8f acc[4];
    #pragma unroll
    for (int t = 0; t < 4; t++) acc[t] = (v8f){0.f,0.f,0.f,0.f,0.f,0.f,0.f,0.f};

    // ---- phase 0: A0 @ B0 ----
    {
        const float* Ar = A0 + (size_t)rowC * K0;
        for (int k0c = 0; k0c < K0; k0c += BK) {
            int KC = (K0 - k0c < BK) ? (K0 - k0c) : BK;
            __syncthreads();                          // LDS WAR vs previous chunk
            stage_b_panel(B0, k0c, KC, n0, N, sB);
            stage_wait();
            for (int kk = 0; kk < KC; kk += 4) {
                v2f a;
                a.x = Ar[k0c + kk + krow];
                a.y = Ar[k0c + kk + krow + 1];
                __builtin_prefetch(Ar + k0c + kk + 16, 0, 0);   // global_prefetch_b8
                #pragma unroll
                for (int t = 0; t < 4; t++) {
                    v2f b;
                    b.x = sB[(kk + krow) * LDB + 16 * t + mr];
                    b.y = sB[(kk + krow + 1) * LDB + 16 * t + mr];
                    acc[t] = __builtin_amdgcn_wmma_f32_16x16x4_f32(
                        false, a, false, b, (short)0, acc[t], false, false);
                }
            }
        }
    }
    // ---- phase 1: A1 @ B1 (optional) ----
    if (A1) {
        const float* Ar = A1 + (size_t)rowC * K1;
        for (int k0c = 0; k0c < K1; k0c += BK) {
            int KC = (K1 - k0c < BK) ? (K1 - k0c) : BK;
            __syncthreads();
            stage_b_panel(B1, k0c, KC, n0, N, sB);
            stage_wait();
            for (int kk = 0; kk < KC; kk += 4) {
                v2f a;
                a.x = Ar[k0c + kk + krow];
                a.y = Ar[k0c + kk + krow + 1];
                __builtin_prefetch(Ar + k0c + kk + 16, 0, 0);
                #pragma unroll
                for (int t = 0; t < 4; t++) {
                    v2f b;
                    b.x = sB[(kk + krow) * LDB + 16 * t + mr];
                    b.y = sB[(kk + krow + 1) * LDB + 16 * t + mr];
                    acc[t] = __builtin_amdgcn_wmma_f32_16x16x4_f32(
                        false, a, false, b, (short)0, acc[t], false, false);
                }
            }
        }
    }

    // Guarded epilogue: only true in-range rows/cols are stored.
    #pragma unroll
    for (int t = 0; t < 4; t++) {
        int col = n0 + 16 * t + mr;
        if (col < N) {
            float bv = bias ? bias[col] : 0.f;
            #pragma unroll
            for (int i = 0; i < 8; i++) {
                int r = m0 + i + 8 * half;      // C/D layout: VGPR i, half-wave -> M=i or i+8
                if (r < M) {
                    size_t idx = (size_t)r * N + col;
                    float v = acc[t][i] + bv;
                    if (pre)  v += pre[idx];
                    if (relu) v = fmaxf(v, 0.f);
                    if (post) v += post[idx];
                    C[idx] = v;
                }
            }
        }
    }
}

// ---------------------------------------------------------------------------
// conv3 GEMV: one wave32 per node. y0 = xo@W3[0]+b3 -> out init; y1 -> scatter src
// ---------------------------------------------------------------------------
__global__ void conv3_kernel(const float* __restrict__ xo, const float* __restrict__ W3,
                             const float* __restrict__ b3, float* __restrict__ out,
                             float* __restrict__ y1, int N, int H) {
    int lane = threadIdx.x & 31;
    int gwarp = blockIdx.x * (blockDim.x >> 5) + (threadIdx.x >> 5);
    int nwarp = gridDim.x * (blockDim.x >> 5);
    for (int i = gwarp; i < N; i += nwarp) {
        const float* r = xo + (size_t)i * H;
        float s0 = 0.f, s1 = 0.f;
        for (int k = lane; k < H; k += 32) {
            float v = r[k];
            s0 += v * W3[k];          // W3 shape (2,H,1) flat
            s1 += v * W3[H + k];
        }
        #pragma unroll
        for (int o = 16; o > 0; o >>= 1) {
            s0 += __shfl_xor(s0, o, 32);
            s1 += __shfl_xor(s1, o, 32);
        }
        if (lane == 0) {
            out[i] = s0 + b3[0];
            y1[i] = s1;
        }
    }
}

// ---------------------------------------------------------------------------
// Edge BCE loss: one wave32 per edge pair; butterfly wave reduction for dots.
// ---------------------------------------------------------------------------
__global__ void loss_kernel(const float* __restrict__ z, const int* __restrict__ pe,
                            const int* __restrict__ ne, float* __restrict__ lacc,
                            int E, int H) {
    __shared__ float sp[8];
    __shared__ float sn[8];
    int wid  = threadIdx.x >> 5;
    int lane = threadIdx.x & 31;
    int gwarp = blockIdx.x * (blockDim.x >> 5) + wid;
    int nwarp = gridDim.x * (blockDim.x >> 5);
    float lp = 0.f, ln = 0.f;
    for (int e = gwarp; e < E; e += nwarp) {
        int s = pe[e], d = pe[E + e];
        const float* zs = z + (size_t)s * H;
        const float* zd = z + (size_t)d * H;
        float dp = 0.f;
        for (int k = lane; k < H; k += 32) dp += zs[k] * zd[k];
        #pragma unroll
        for (int o = 16; o > 0; o >>= 1) dp += __shfl_xor(dp, o, 32);

        s = ne[e]; d = ne[E + e];
        zs = z + (size_t)s * H;
        zd = z + (size_t)d * H;
        float dn = 0.f;
        for (int k = lane; k < H; k += 32) dn += zs[k] * zd[k];
        #pragma unroll
        for (int o = 16; o > 0; o >>= 1) dn += __shfl_xor(dn, o, 32);

        if (lane == 0) {
            lp += logf(1.f / (1.f + expf(-dp)) + 1e-15f);
            ln += logf(1.f - 1.f / (1.f + expf(-dn)) + 1e-15f);
        }
    }
    if (lane == 0) { sp[wid] = lp; sn[wid] = ln; }
    __syncthreads();
    if (threadIdx.x == 0) {
        float a = 0.f, b = 0.f;
        int nw = blockDim.x >> 5;
        for (int i = 0; i < nw; i++) { a += sp[i]; b += sn[i]; }
        atomicAdd(&lacc[0], a);
        atomicAdd(&lacc[1], b);
    }
}

__global__ void finalize_kernel(const float* lacc, const float* c1, const float* c2,
                                float* out, int N, float invE) {
    out[N]     = -(lacc[0] + lacc[1]) * invE;   // pos_loss + neg_loss
    out[N + 1] = c1[0];
    out[N + 2] = c2[0];
}

// ---------------------------------------------------------------------------
extern "C" void kernel_launch(void* const* d_in, const int* in_sizes, int n_in,
                              void* d_out, int out_size, void* d_ws, size_t ws_size,
                              hipStream_t stream) {
    const float* x     = (const float*)d_in[0];
    const int*   ei    = (const int*)d_in[1];
    const int*   nei   = (const int*)d_in[2];
    const float* W1    = (const float*)d_in[3];
    const float* b1    = (const float*)d_in[4];
    const float* W2    = (const float*)d_in[5];
    const float* b2    = (const float*)d_in[6];
    const float* W3    = (const float*)d_in[7];
    const float* b3    = (const float*)d_in[8];
    const float* linW1 = (const float*)d_in[9];
    const float* linb1 = (const float*)d_in[10];
    const float* linW2 = (const float*)d_in[11];
    const float* linb2 = (const float*)d_in[12];
    const float* c1    = (const float*)d_in[13];
    const float* c2    = (const float*)d_in[14];

    const int F  = 64;
    const int N  = in_sizes[0] / F;          // 13627
    const int E  = in_sizes[1] / 2;          // 504378
    const int H1 = in_sizes[3] / (2 * F);    // 300
    const int H2 = in_sizes[5] / (2 * H1);   // 100

    float* ws = (float*)d_ws;
    size_t off = 0;
    float* deg  = ws + off; off += (size_t)N;          // becomes dinv in place
    float* norm = ws + off; off += (size_t)E;
    float* tx   = ws + off; off += (size_t)N * F;      // segsum(norm*x[src])
    float* h    = ws + off; off += (size_t)N * H1;
    float* g1   = ws + off; off += (size_t)N * H2;     // h @ W2[1]
    float* t2   = ws + off; off += (size_t)N * H2;     // segsum(norm*g1[src])
    float* x1   = ws + off; off += (size_t)N * H2;
    float* xo   = ws + off; off += (size_t)N * H2;
    float* zb   = ws + off; off += (size_t)N * H2;
    float* y1   = ws + off; off += (size_t)N;
    float* lacc = ws + off; off += 2;
    (void)ws_size; (void)n_in; (void)out_size;

    float* out = (float*)d_out;
    auto cdiv = [](int a, int b) { return (a + b - 1) / b; };

    // 1. degree -> dinv -> edge norms
    zero_f32<<<cdiv(N, 256), 256, 0, stream>>>(deg, (size_t)N);
    degree_kernel<<<cdiv(E, 256), 256, 0, stream>>>(ei, deg, E);
    dinv_kernel<<<cdiv(N, 256), 256, 0, stream>>>(deg, N);
    norm_kernel<<<cdiv(E, 256), 256, 0, stream>>>(ei, deg, norm, E);

    // 2. conv1: h = relu(x@W1[0] + segsum(norm*x[src])@W1[1] + b1)
    zero_f32<<<cdiv(N * F, 256), 256, 0, stream>>>(tx, (size_t)N * F);
    scatter_axpy<<<2048, 256, 0, stream>>>(x, ei, norm, tx, E, F);
    {
        dim3 g(cdiv(H1, 64), cdiv(N, 128));
        gemm_wmma_f32<<<g, 256, 0, stream>>>(x, W1, F, tx, W1 + (size_t)F * H1, F,
                                             b1, nullptr, nullptr, h, N, H1, 1);
    }

    // 3. conv2 (GEMM-then-scatter: segsum(norm*h[src])@W == segsum(norm*(h@W)[src]))
    dim3 g2(cdiv(H2, 64), cdiv(N, 128));
    gemm_wmma_f32<<<g2, 256, 0, stream>>>(h, W2 + (size_t)H1 * H2, H1,
                                          nullptr, nullptr, 0,
                                          nullptr, nullptr, nullptr, g1, N, H2, 0);
    zero_f32<<<cdiv(N * H2, 256), 256, 0, stream>>>(t2, (size_t)N * H2);
    scatter_axpy<<<2048, 256, 0, stream>>>(g1, ei, norm, t2, E, H2);
    gemm_wmma_f32<<<g2, 256, 0, stream>>>(h, W2, H1, nullptr, nullptr, 0,
                                          b2, t2, nullptr, x1, N, H2, 1);

    // 4. skip branches: xo = x1 + relu(x@linW1+linb1) ; z = x1 + relu(x@linW2+linb2)
    gemm_wmma_f32<<<g2, 256, 0, stream>>>(x, linW1, F, nullptr, nullptr, 0,
                                          linb1, nullptr, x1, xo, N, H2, 1);
    gemm_wmma_f32<<<g2, 256, 0, stream>>>(x, linW2, F, nullptr, nullptr, 0,
                                          linb2, nullptr, x1, zb, N, H2, 1);

    // 5. edge-sampling BCE loss over z
    zero_f32<<<1, 32, 0, stream>>>(lacc, 2);
    loss_kernel<<<1024, 256, 0, stream>>>(zb, ei, nei, lacc, E, H2);

    // 6. conv3 (GEMV, width-1 scatter)
    conv3_kernel<<<cdiv(N * 32, 256), 256, 0, stream>>>(xo, W3, b3, out, y1, N, H2);
    scatter1<<<cdiv(E, 256), 256, 0, stream>>>(y1, ei, norm, out, E);

    // 7. scalars: r_loss, c1, c2
    finalize_kernel<<<1, 1, 0, stream>>>(lacc, c1, c2, out, N, 1.f / (float)E);
}